// NGNNConv_78572131713430
// MI455X (gfx1250) — compile-verified
//
#include <hip/hip_runtime.h>

// Problem constants (match reference setup_inputs()).
constexpr int N    = 1024;
constexpr int IND  = 32;
constexpr int OUTD = 32;
constexpr int E    = 8192;   // power of two -> bitonic-sort friendly

typedef __attribute__((ext_vector_type(2))) float        v2f;
typedef __attribute__((ext_vector_type(8))) float        v8f;
typedef __attribute__((ext_vector_type(4))) unsigned int v4u;
typedef __attribute__((ext_vector_type(4))) int          v4i;
typedef __attribute__((ext_vector_type(8))) int          v8i;

// ---------------------------------------------------------------------------
// Kernel 1: deterministic CSR build.
// Bitonic sort of keys (row<<13 | edge_idx) in LDS: stable grouping by
// destination row with original edge order preserved -> deterministic float
// accumulation order in the main kernel. Offsets via binary search.
// ---------------------------------------------------------------------------
__global__ void __launch_bounds__(1024) ngnn_build_csr(
    const int* __restrict__ erow, const int* __restrict__ ecol,
    int* __restrict__ csr_off, int* __restrict__ csr_col) {
  __shared__ int keys[E];
  const int tid = threadIdx.x;

  for (int i = tid; i < E; i += 1024) keys[i] = (erow[i] << 13) | i;
  __syncthreads();

  for (int k = 2; k <= E; k <<= 1) {
    for (int j = k >> 1; j > 0; j >>= 1) {
      for (int i = tid; i < E; i += 1024) {
        const int p = i ^ j;
        if (p > i) {
          const int a = keys[i];
          const int b = keys[p];
          const bool up = ((i & k) == 0);
          if ((a > b) == up) { keys[i] = b; keys[p] = a; }
        }
      }
      __syncthreads();
    }
  }

  for (int i = tid; i < E; i += 1024) csr_col[i] = ecol[keys[i] & (E - 1)];
  // csr_off[r] = first sorted index with row >= r  (csr_off[N] == E).
  for (int r = tid; r <= N; r += 1024) {
    int lo = 0, hi = E;
    while (lo < hi) {
      const int mid = (lo + hi) >> 1;
      if ((keys[mid] >> 13) < r) lo = mid + 1; else hi = mid;
    }
    csr_off[r] = lo;
  }
}

// ---------------------------------------------------------------------------
// Kernel 2: fused gather + masked-linear via WMMA, with TDM panel staging.
// Block b handles subgraph i = b. X[i,:,:] (128 KB) is DMA'd into LDS by the
// Tensor Data Mover (one tensor_load_to_lds, TENSORcnt-tracked) while the
// waves stage mask[i,:] and preload W/b. Each wave owns 16-row j-tiles:
//   lane L (jl = L&15, hl = L>>4) accumulates S for destination j = jb+jl in
//   the V_WMMA_F32_16X16X4_F32 A-layout (hl=0: d = 4c,4c+1; hl=1: d = 4c+2,
//   4c+3), then 8 chained fp32 WMMAs per 16x16 output tile, 2 tiles (OUTD=32).
// ---------------------------------------------------------------------------
__global__ void __launch_bounds__(256) ngnn_fused(
    const float* __restrict__ X, const int* __restrict__ mask,
    const float* __restrict__ W, const float* __restrict__ bvec,
    const int* __restrict__ csr_off, const int* __restrict__ csr_col,
    float* __restrict__ out) {
  extern __shared__ float smem[];
  float* Xp    = smem;            // N*IND floats = 128 KB (TDM destination)
  float* maskp = smem + N * IND;  // N floats

  const int i   = blockIdx.x;
  const int tid = threadIdx.x;

  // ---- Issue the TDM DMA of X[i,:,:] -> LDS from wave 0 only (uniform). ----
  if (tid < 32) {
    const unsigned long long gaddr =
        (unsigned long long)(const void*)(X + (size_t)i * (N * IND));
    const unsigned int lds_addr = (unsigned int)(unsigned long long)(void*)Xp;

    // D# group 0: count=1, lds_addr, global_addr[56:0], type=2 (bits 127:126).
    v4u g0;
    g0.x = 1u;                               // count=1, no gather/restore
    g0.y = lds_addr;                         // lds_addr (bytes)
    g0.z = (unsigned int)gaddr;              // global_addr[31:0]
    g0.w = ((unsigned int)(gaddr >> 32) & 0x01FFFFFFu) | (2u << 30);

    // D# group 1: data_size=4B (code 2); 1D tile: tensor_dim0 = tile_dim0 =
    // 32768 elements; tensor_dim1=1; tile_dim1/2 unused; stride0=32768.
    v8i g1;
    g1[0] = 0x00020000;         // workgroup_mask=0, data_size=2, no flags
    g1[1] = (int)0x80000000u;   // tensor_dim0[15:0]=0x8000 in bits 63:48
    g1[2] = 0x00010000;         // tensor_dim0[31:16]=0, tensor_dim1[15:0]=1
    g1[3] = (int)0x80000000u;   // tensor_dim1[31:16]=0, tile_dim0=0x8000
    g1[4] = 0;                  // tile_dim1=0 (unused), tile_dim2=0
    g1[5] = 32768;              // tensor_dim0_stride[31:0]
    g1[6] = 0;                  // stride0[47:32]=0, tensor_dim1_stride[15:0]=0
    g1[7] = 0;                  // tensor_dim1_stride[47:16]=0

    const v4i gz4 = {0, 0, 0, 0};              // groups 2/3 (<=2D tensor)
    const v8i gz8 = {0, 0, 0, 0, 0, 0, 0, 0};  // extra operand (clang-23 form)
    __builtin_amdgcn_tensor_load_to_lds(g0, g1, gz4, gz4, gz8, 0);
  }

  // ---- Overlap with the DMA: stage mask row, preload W/b. ----
  const int* mi = mask + (size_t)i * N;
  for (int idx = tid; idx < N; idx += 256) maskp[idx] = (float)mi[idx];

  const int lane = tid & 31;
  const int wv   = tid >> 5;
  const int jl   = lane & 15;
  const int hl   = lane >> 4;

  // Preload B (= W) in the 16x16x4 B-layout:
  // VGPR0: K=0 (lanes 0-15) / K=2 (lanes 16-31); VGPR1: K=1 / K=3.
  v2f Bw[2][8];
#pragma unroll
  for (int nt = 0; nt < 2; ++nt) {
#pragma unroll
    for (int c = 0; c < 8; ++c) {
      const int d0 = 4 * c + 2 * hl;
      const int n  = jl + 16 * nt;
      Bw[nt][c].x = W[d0 * OUTD + n];
      Bw[nt][c].y = W[(d0 + 1) * OUTD + n];
    }
  }
  const float bn0 = bvec[jl];
  const float bn1 = bvec[jl + 16];

  // Wave 0 waits for the DMA, then the barrier publishes the panel to all.
  if (tid < 32) __builtin_amdgcn_s_wait_tensorcnt(0);
  __syncthreads();

  float* outi = out + (size_t)i * (N * OUTD);

  for (int tt = wv; tt < N / 16; tt += 8) {
    const int jb = tt * 16;
    const int j  = jb + jl;
    const int o0 = csr_off[j];
    const int o1 = csr_off[j + 1];

    float S[16];
#pragma unroll
    for (int q = 0; q < 16; ++q) S[q] = 0.f;
    float cc = 0.f;

    // Gather masked neighbor rows from the LDS panel (deterministic order).
    int t = 0;
    while (__any((o0 + t) < o1)) {
      if ((o0 + t) < o1) {
        const int   k  = csr_col[o0 + t];
        const float mf = maskp[k];
        cc += mf;
        const float* xr = Xp + (k << 5) + (hl << 1);
#pragma unroll
        for (int c = 0; c < 8; ++c) {
          const v2f v = *reinterpret_cast<const v2f*>(xr + 4 * c);  // ds_load_b64
          S[2 * c]     += mf * v.x;
          S[2 * c + 1] += mf * v.y;
        }
      }
      ++t;
    }

    // D = S @ W + c*b  via chained fp32 WMMAs (EXEC all-ones here).
#pragma unroll
    for (int nt = 0; nt < 2; ++nt) {
      const float bn = nt ? bn1 : bn0;
      v8f acc;
#pragma unroll
      for (int r = 0; r < 8; ++r) {
        // C/D row M = r + 8*hl -> fetch that destination's edge-mask count.
        const float cr = __shfl(cc, r + 8 * hl, 32);
        acc[r] = cr * bn;
      }
#pragma unroll
      for (int c = 0; c < 8; ++c) {
        v2f a;
        a.x = S[2 * c];
        a.y = S[2 * c + 1];
        acc = __builtin_amdgcn_wmma_f32_16x16x4_f32(
            /*neg_a=*/false, a, /*neg_b=*/false, Bw[nt][c],
            /*c_mod=*/(short)0, acc, /*reuse_a=*/false, /*reuse_b=*/false);
      }
#pragma unroll
      for (int r = 0; r < 8; ++r) {
        const int jr = jb + r + 8 * hl;
        outi[jr * OUTD + jl + 16 * nt] = maskp[jr] * acc[r];
      }
    }
  }
}

// ---------------------------------------------------------------------------
extern "C" void kernel_launch(void* const* d_in, const int* in_sizes, int n_in,
                              void* d_out, int out_size, void* d_ws, size_t ws_size,
                              hipStream_t stream) {
  (void)in_sizes; (void)n_in; (void)out_size; (void)ws_size;
  const float* X    = (const float*)d_in[0];
  const int*   mask = (const int*)d_in[1];
  const int*   erow = (const int*)d_in[2];
  const int*   ecol = (const int*)d_in[3];
  const float* W    = (const float*)d_in[4];
  const float* b    = (const float*)d_in[5];
  float*       out  = (float*)d_out;

  int* csr_off = (int*)d_ws;           // N+1 ints
  int* csr_col = ((int*)d_ws) + 1056;  // E ints (offset keeps 128B alignment)

  ngnn_build_csr<<<1, 1024, 0, stream>>>(erow, ecol, csr_off, csr_col);

  const size_t lds_bytes = (size_t)(N * IND + N) * sizeof(float);  // 132 KB
  ngnn_fused<<<N, 256, lds_bytes, stream>>>(X, mask, W, b, csr_off, csr_col, out);
}